// sensitivity_prediction_2_11716670783534
// MI455X (gfx1250) — compile-verified
//
#include <hip/hip_runtime.h>

typedef float v2f __attribute__((ext_vector_type(2)));
typedef float v8f __attribute__((ext_vector_type(8)));

#define NE        8192
#define BATCH     64
#define NNZ       (NE * 32)
#define NEG_SLOPE 0.001f

// ---------------------------------------------------------------------------
// GEMM: H[b, i] = sum_k X[b, k] * W[i, k]   (M=64, N=8192, K=8192), f32 WMMA.
// Wave tiling: each wave computes a 64x16 tile of H via 4 accumulators of
// V_WMMA_F32_16X16X4_F32. 8 waves/block -> 128 columns/block, 64 blocks.
//
// A (16x4 f32) lane layout: lanes 0-15 = rows M, VGPR0 holds K=2*half+0,
// VGPR1 holds K=2*half+1 (half = lane>>4). B (4x16) mirrors this with
// lane%16 = N column. Both reduce to a contiguous float2 load at
// base + k + 2*half, which keeps the inner loop pure b64 loads + WMMA.
// ---------------------------------------------------------------------------
__global__ __launch_bounds__(256) void gemm_xwT_wmma(
    const float* __restrict__ X,   // [BATCH, NE]
    const float* __restrict__ W,   // [NE, NE] row-major (row i = weights of out i)
    float* __restrict__ H)         // [BATCH, NE]
{
  const int lane = threadIdx.x & 31;
  const int wave = threadIdx.x >> 5;        // 0..7
  const int lm   = lane & 15;
  const int half = lane >> 4;               // 0 or 1

  const int ncol = blockIdx.x * 128 + wave * 16 + lm;   // this lane's N column

  const float* pb  = W + (size_t)ncol * NE + 2 * half;  // B operand: W row ncol
  const float* pa0 = X + (size_t)(lm)      * NE + 2 * half;
  const float* pa1 = pa0 + (size_t)16 * NE;
  const float* pa2 = pa0 + (size_t)32 * NE;
  const float* pa3 = pa0 + (size_t)48 * NE;

  v8f c0 = {}, c1 = {}, c2 = {}, c3 = {};

  for (int k = 0; k < NE; k += 32) {
    // keep the HBM stream of W ahead of the math (global_prefetch_b8)
    __builtin_prefetch(pb + k + 2048, 0, 1);
#pragma unroll
    for (int kk = 0; kk < 32; kk += 4) {
      v2f b  = *(const v2f*)(pb  + k + kk);
      v2f a0 = *(const v2f*)(pa0 + k + kk);
      v2f a1 = *(const v2f*)(pa1 + k + kk);
      v2f a2 = *(const v2f*)(pa2 + k + kk);
      v2f a3 = *(const v2f*)(pa3 + k + kk);
      c0 = __builtin_amdgcn_wmma_f32_16x16x4_f32(false, a0, false, b, (short)0, c0, false, false);
      c1 = __builtin_amdgcn_wmma_f32_16x16x4_f32(false, a1, false, b, (short)0, c1, false, false);
      c2 = __builtin_amdgcn_wmma_f32_16x16x4_f32(false, a2, false, b, (short)0, c2, false, false);
      c3 = __builtin_amdgcn_wmma_f32_16x16x4_f32(false, a3, false, b, (short)0, c3, false, false);
    }
  }

  // C/D 16x16 f32 layout: VGPR r -> M = r + 8*half (lanes 16-31 hold M=8..15),
  // N = lane % 16.
  const int mbase = 8 * half;
#pragma unroll
  for (int r = 0; r < 8; ++r) {
    H[(size_t)( 0 + mbase + r) * NE + ncol] = c0[r];
    H[(size_t)(16 + mbase + r) * NE + ncol] = c1[r];
    H[(size_t)(32 + mbase + r) * NE + ncol] = c2[r];
    H[(size_t)(48 + mbase + r) * NE + ncol] = c3[r];
  }
}

// ---------------------------------------------------------------------------
// out must start at zero (harness poisons d_out).
// ---------------------------------------------------------------------------
__global__ void zero_out_kernel(float* __restrict__ out, int n) {
  int i = blockIdx.x * blockDim.x + threadIdx.x;
  if (i < n) out[i] = 0.0f;
}

// ---------------------------------------------------------------------------
// Sparse scatter: out[b, col[n]] += H[b, row[n]] * w[n].
// 64 consecutive threads share one nnz entry (one index/weight fetch, 64
// batches in parallel); H (2 MB) and out (2 MB) are L2-resident so the
// 16.8M global_atomic_add_f32 land in L2.
// ---------------------------------------------------------------------------
__global__ __launch_bounds__(256) void scatter_spmm_kernel(
    const float*      __restrict__ H,    // [BATCH, NE]
    const float*      __restrict__ sw,   // [NNZ]
    const long long*  __restrict__ idx,  // [2, NNZ] int64 (rows then cols)
    float*            __restrict__ out)  // [BATCH, NE]
{
  const int b   = threadIdx.x & 63;      // batch index
  const int sub = threadIdx.x >> 6;      // 0..3 nnz slots per block iter

  for (int n = blockIdx.x * 4 + sub; n < NNZ; n += gridDim.x * 4) {
    const int   row = (int)idx[n];
    const int   col = (int)idx[NNZ + n];
    const float w   = sw[n];
    const float v   = H[(size_t)b * NE + row] * w;
    atomicAdd(&out[(size_t)b * NE + col], v);
  }
}

// ---------------------------------------------------------------------------
// LeakyReLU (slope 0.001) after all scatters complete.
// ---------------------------------------------------------------------------
__global__ void leaky_relu_kernel(float* __restrict__ out, int n) {
  int i = blockIdx.x * blockDim.x + threadIdx.x;
  if (i < n) {
    float v = out[i];
    out[i] = (v >= 0.0f) ? v : NEG_SLOPE * v;
  }
}

extern "C" void kernel_launch(void* const* d_in, const int* in_sizes, int n_in,
                              void* d_out, int out_size, void* d_ws, size_t ws_size,
                              hipStream_t stream) {
  const float*     x   = (const float*)d_in[0];          // [64, 8192]
  const float*     W   = (const float*)d_in[1];          // [8192, 8192]
  const float*     sw  = (const float*)d_in[2];          // [NNZ]
  const long long* idx = (const long long*)d_in[3];      // [2, NNZ] int64
  float*           out = (float*)d_out;                  // [64, 8192]
  float*           H   = (float*)d_ws;                   // 64*8192*4 = 2 MB scratch

  const int out_elems = BATCH * NE;                      // 524288

  // 1) zero the (poisoned) output accumulator
  zero_out_kernel<<<(out_elems + 255) / 256, 256, 0, stream>>>(out, out_elems);

  // 2) dense layer on the f32 WMMA path: 64 blocks x 8 waves cover N=8192
  gemm_xwT_wmma<<<NE / 128, 256, 0, stream>>>(x, W, H);

  // 3) gather/scale/scatter-add over the static COO pattern
  scatter_spmm_kernel<<<1024, 256, 0, stream>>>(H, sw, idx, out);

  // 4) leaky ReLU in place
  leaky_relu_kernel<<<(out_elems + 255) / 256, 256, 0, stream>>>(out, out_elems);
}